// ROIAlign_19086834663990
// MI455X (gfx1250) — compile-verified
//
#include <hip/hip_runtime.h>

// ROIAlign forward for MI455X (gfx1250, wave32).
// Shapes fixed by the reference: feat[2,256,200,272] f32, rois[K,5] f32,
// out[K,256,7,7] f32.  Memory-gather bound: feature map (111 MB) is L2-
// resident (192 MB L2); output (50 MB) is streamed with NT stores.

#define POOLED_H 7
#define POOLED_W 7
#define SR 2
#define C_ 256
#define H_ 200
#define W_ 272
#define SCALE 0.25f
#define HW_ (H_ * W_)
#define CHW_ (C_ * HW_)
#define ELEMS_PER_ROI (C_ * POOLED_H * POOLED_W)   // 12544
#define BLOCKS_PER_ROI (ELEMS_PER_ROI / 256)       // 49 (exact)

__global__ __launch_bounds__(256) void roialign_fwd_kernel(
    const float* __restrict__ feat,
    const float* __restrict__ rois,
    float* __restrict__ out,
    int K)
{
    const int blk = blockIdx.x;
    const int k   = blk / BLOCKS_PER_ROI;          // ROI index: block-uniform
    const int r   = blk - k * BLOCKS_PER_ROI;

    // ---- wave-uniform ROI geometry (compiler keeps this scalar) ----
    const float* roi = rois + k * 5;
    const int   b       = (int)roi[0];
    const float start_x = roi[1] * SCALE;
    const float start_y = roi[2] * SCALE;
    const float roi_w   = fmaxf(roi[3] * SCALE - start_x, 1.0f);
    const float roi_h   = fmaxf(roi[4] * SCALE - start_y, 1.0f);
    const float bin_w   = roi_w * (1.0f / POOLED_W);
    const float bin_h   = roi_h * (1.0f / POOLED_H);
    const float* fb = feat + b * CHW_;

    // Prefetch next ROI descriptor into cache (gfx1250 global_prefetch_b8).
    if (k + 1 < K) {
        __builtin_prefetch(roi + 5, 0, 1);
    }

    // ---- per-thread output coordinate ----
    const int e   = r * 256 + (int)threadIdx.x;    // [0, 12544)
    const int c   = e / (POOLED_H * POOLED_W);
    const int bin = e - c * (POOLED_H * POOLED_W);
    const int ph  = bin / POOLED_W;
    const int pw  = bin - ph * POOLED_W;
    const float* fc = fb + c * HW_;

    float acc = 0.0f;
    #pragma unroll
    for (int iy = 0; iy < SR; ++iy) {
        const float y  = start_y + bin_h * ((float)ph + ((float)iy + 0.5f) * (1.0f / SR));
        const bool  vy = (y >= -1.0f) && (y <= (float)H_);
        const float yc = fmaxf(y, 0.0f);
        int   yl = min((int)yc, H_ - 1);           // yc >= 0 so trunc == floor
        int   yh = min(yl + 1, H_ - 1);
        float ly = (yl >= H_ - 1) ? 0.0f : (yc - (float)yl);
        float hy = 1.0f - ly;

        #pragma unroll
        for (int ix = 0; ix < SR; ++ix) {
            const float x  = start_x + bin_w * ((float)pw + ((float)ix + 0.5f) * (1.0f / SR));
            const bool  vx = (x >= -1.0f) && (x <= (float)W_);
            const float xc = fmaxf(x, 0.0f);
            int   xl = min((int)xc, W_ - 1);
            int   xh = min(xl + 1, W_ - 1);
            float lx = (xl >= W_ - 1) ? 0.0f : (xc - (float)xl);
            float hx = 1.0f - lx;

            const float* r0 = fc + yl * W_;
            const float* r1 = fc + yh * W_;
            // 4 independent gathers; unrolled loop gives 16 in flight before
            // the s_wait_loadcnt.
            const float v00 = r0[xl];
            const float v01 = r0[xh];
            const float v10 = r1[xl];
            const float v11 = r1[xh];

            const float s = hy * (hx * v00 + lx * v01) + ly * (hx * v10 + lx * v11);
            acc += (vy && vx) ? s : 0.0f;
        }
    }

    // Mean over SR*SR samples; NT store keeps the L2 free for the feature map.
    __builtin_nontemporal_store(acc * (1.0f / (SR * SR)), out + (size_t)blk * 256 + threadIdx.x);
}

extern "C" void kernel_launch(void* const* d_in, const int* in_sizes, int n_in,
                              void* d_out, int out_size, void* d_ws, size_t ws_size,
                              hipStream_t stream) {
    const float* feat = (const float*)d_in[0];   // [B, C, H, W] f32
    const float* rois = (const float*)d_in[1];   // [K, 5] f32
    float* out = (float*)d_out;                  // [K, C, 7, 7] f32

    const int K = in_sizes[1] / 5;
    const int grid = K * BLOCKS_PER_ROI;         // one block per (roi, 256-elem slab)

    roialign_fwd_kernel<<<grid, 256, 0, stream>>>(feat, rois, out, K);
}